// IRadon_82282983457007
// MI455X (gfx1250) — compile-verified
//
#include <hip/hip_runtime.h>
#include <math.h>

typedef __attribute__((ext_vector_type(2))) float v2f;
typedef __attribute__((ext_vector_type(8))) float v8f;

#define H   1024
#define NA  180
#define NB  16
#define WIN 48

// ---------------------------------------------------------------------------
// Kernel 1: ramp filter as implicit Toeplitz GEMM using V_WMMA_F32_16X16X4_F32
//   xf_t[b,j,n] = sum_k g(n-k) * x[b,k,j]    (output stored TRANSPOSED)
// One wave computes a 64x16 panel (4 row-subtiles share each B fragment).
// The Toeplitz kernel g(m) is stored REVERSED in LDS so each A fragment is an
// ascending consecutive pair (no swap movs). Out-of-range columns are handled
// by index clamping (no exec-mask predication); their garbage results are
// simply not stored. C/D layout: VGPR v holds row M = v + (lane<16?0:8),
// col N = lane&15  ->  8 contiguous floats per lane in transposed layout.
// ---------------------------------------------------------------------------
__global__ __launch_bounds__(128) void k_filter_gemm(const float* __restrict__ x,
                                                     float* __restrict__ xf_t) {
  __shared__ float tab[2048];          // tab[i] = g(1023 - i)
  const int tid = threadIdx.x;
  for (int i = tid; i < 2046; i += 128) {
    int m = 1023 - i;
    float g;
    if (m == 0)       g = 0.5f;
    else if (m & 1) { float fm = (float)m; g = -0.20264236728467558f / (fm * fm); }
    else              g = 0.0f;
    tab[i] = g;
  }
  __syncthreads();

  const int lane  = tid & 31;
  const int wv    = tid >> 5;
  const int jt    = blockIdx.x * 4 + wv;   // column tile (0..11), 12*16 >= 180
  const int jcol  = jt * 16 + (lane & 15);
  const int mbase = blockIdx.y * 64;       // 64 output rows per wave
  const int b     = blockIdx.z;

  const int krow2 = (lane >> 4) << 1;      // 0 for lanes 0-15, 2 for 16-31
  const int jc    = min(jcol, NA - 1);     // clamp: garbage cols never stored
  const float* xcol = x + (size_t)b * H * NA + jc;

  v8f acc0 = {}, acc1 = {}, acc2 = {}, acc3 = {};
  const int n0 = mbase + (lane & 15);
  // reversed-table index: g(n0 - K) = tab[1023 - n0 + K]
  const int ribase = 1023 - n0;

  for (int k = 0; k < H; k += 4) {
    const int K = k + krow2;
    v2f bf;
    bf.x = xcol[(size_t)K * NA];
    bf.y = xcol[(size_t)K * NA + NA];

    const int ri = ribase + K;             // ascending consecutive pairs
    v2f a0, a1, a2, a3;
    a0.x = tab[ri];      a0.y = tab[ri + 1];
    a1.x = tab[ri - 16]; a1.y = tab[ri - 15];
    a2.x = tab[ri - 32]; a2.y = tab[ri - 31];
    a3.x = tab[ri - 48]; a3.y = tab[ri - 47];

    acc0 = __builtin_amdgcn_wmma_f32_16x16x4_f32(false, a0, false, bf, (short)0, acc0, false, false);
    acc1 = __builtin_amdgcn_wmma_f32_16x16x4_f32(false, a1, false, bf, (short)0, acc1, false, false);
    acc2 = __builtin_amdgcn_wmma_f32_16x16x4_f32(false, a2, false, bf, (short)0, acc2, false, false);
    acc3 = __builtin_amdgcn_wmma_f32_16x16x4_f32(false, a3, false, bf, (short)0, acc3, false, false);
  }

  if (jcol < NA) {
    const int rowoff = (lane >> 4) << 3;   // 0 or 8
    float* dst = xf_t + ((size_t)b * NA + jcol) * H + mbase + rowoff;
    // 32B-aligned (mbase%64==0, rowoff in {0,8}); 2x global_store_b128 each
    *reinterpret_cast<v8f*>(dst)      = acc0;
    *reinterpret_cast<v8f*>(dst + 16) = acc1;
    *reinterpret_cast<v8f*>(dst + 32) = acc2;
    *reinterpret_cast<v8f*>(dst + 48) = acc3;
  }
}

// ---------------------------------------------------------------------------
// Kernel 2: angle resample (A->A lerp with edge-zero weights). Input and
// output are both (b, angle, detector), so reads and writes are coalesced.
// ---------------------------------------------------------------------------
__global__ __launch_bounds__(256) void k_resample_t(const float* __restrict__ xf_t,
                                                    float* __restrict__ xat) {
  const int n = blockIdx.x * 256 + threadIdx.x;
  const int a = blockIdx.y;
  const int b = blockIdx.z;

  const float ix  = (float)a * (180.0f / 179.0f) - 0.5f;
  const float i0f = floorf(ix);
  const int   i0  = (int)i0f;
  const float fx  = ix - i0f;
  const int   i1  = i0 + 1;
  const float w0  = (i0 >= 0 && i0 < NA) ? 1.0f - fx : 0.0f;
  const float w1  = (i1 >= 0 && i1 < NA) ? fx        : 0.0f;
  const int   i0c = min(max(i0, 0), NA - 1);
  const int   i1c = min(max(i1, 0), NA - 1);

  const float* base = xf_t + (size_t)b * NA * H + n;
  xat[((size_t)b * NA + a) * H + n] =
      base[(size_t)i0c * H] * w0 + base[(size_t)i1c * H] * w1;
}

// ---------------------------------------------------------------------------
// Kernel 3: backprojection. One block = 32x32 pixel tile. For each of the 180
// angles we stage the 48-detector window the tile can touch into LDS
// (geometry bounds the span to <=44 samples), then every pixel lerps from LDS.
// ---------------------------------------------------------------------------
__global__ __launch_bounds__(256) void k_backproject(const float* __restrict__ xat,
                                                     float* __restrict__ out) {
  __shared__ float win[NA * WIN];
  __shared__ float lc[NA], ls[NA];
  __shared__ int   lb[NA];

  const int tid = threadIdx.x;
  const int tx0 = blockIdx.x * 32;
  const int ty0 = blockIdx.y * 32;
  const int b   = blockIdx.z;
  const float inv1023 = 2.0f / 1023.0f;

  if (tid < NA) {
    float s, c;
    __sincosf((float)tid * 0.017453292519943295f, &s, &c);
    lc[tid] = c; ls[tid] = s;
    const float x0u = tx0 * inv1023 - 1.0f;
    const float x1u = (tx0 + 31) * inv1023 - 1.0f;
    const float y0u = ty0 * inv1023 - 1.0f;
    const float y1u = (ty0 + 31) * inv1023 - 1.0f;
    const float Tmin  = fminf(x0u * c, x1u * c) + fminf(-y0u * s, -y1u * s);
    const float iymin = Tmin * 512.0f + 511.5f;     // iy = ((T+1)*H - 1)/2
    lb[tid] = (int)floorf(iymin) - 1;
  }
  __syncthreads();

  const float* xb = xat + (size_t)b * NA * H;
  for (int idx = tid; idx < NA * WIN; idx += 256) {
    const int a = idx / WIN;
    const int w = idx - a * WIN;
    int j = lb[a] + w;
    j = min(max(j, 0), H - 1);
    win[idx] = xb[(size_t)a * H + j];
  }
  __syncthreads();

  const int lx = tid & 31;
  const int ly = tid >> 5;
  const int px = tx0 + lx;
  const float xg = px * inv1023 - 1.0f;

  float yg[4], acc[4];
  int py[4];
  #pragma unroll
  for (int p = 0; p < 4; ++p) {
    py[p]  = ty0 + ly + 8 * p;
    yg[p]  = py[p] * inv1023 - 1.0f;
    acc[p] = 0.0f;
  }

  for (int a = 0; a < NA; ++a) {
    const float c  = lc[a];
    const float s  = ls[a];
    const int   bj = lb[a];
    const float* wa = &win[a * WIN];
    #pragma unroll
    for (int p = 0; p < 4; ++p) {
      const float T   = xg * c - yg[p] * s;
      const float iy  = T * 512.0f + 511.5f;
      const float j0f = floorf(iy);
      const float fy  = iy - j0f;
      const int   j0  = (int)j0f;
      const int   j1  = j0 + 1;
      const float u0  = ((unsigned)j0 < (unsigned)H) ? 1.0f - fy : 0.0f;
      const float u1  = ((unsigned)j1 < (unsigned)H) ? fy        : 0.0f;
      const int   l0  = j0 - bj;                    // in [1, 46] by geometry
      acc[p] += wa[l0] * u0 + wa[l0 + 1] * u1;
    }
  }

  const float scale = 0.008726646259971648f;        // pi / 360
  float* ob = out + (size_t)b * H * H;
  #pragma unroll
  for (int p = 0; p < 4; ++p) {
    const float r2 = xg * xg + yg[p] * yg[p];
    ob[(size_t)py[p] * H + px] = (r2 <= 1.0f) ? acc[p] * scale : 0.0f;
  }
}

// ---------------------------------------------------------------------------
extern "C" void kernel_launch(void* const* d_in, const int* in_sizes, int n_in,
                              void* d_out, int out_size, void* d_ws, size_t ws_size,
                              hipStream_t stream) {
  (void)in_sizes; (void)n_in; (void)out_size; (void)ws_size;
  const float* x = (const float*)d_in[0];
  float* out  = (float*)d_out;
  float* xf_t = (float*)d_ws;                       // (16,180,1024) filtered^T
  float* xat  = xf_t + (size_t)NB * NA * H;         // (16,180,1024) resampled

  k_filter_gemm<<<dim3(3, 16, NB), 128, 0, stream>>>(x, xf_t);
  k_resample_t<<<dim3(H / 256, NA, NB), 256, 0, stream>>>(xf_t, xat);
  k_backproject<<<dim3(32, 32, NB), 256, 0, stream>>>(xat, out);
}